// EntropyGatedSlotModel_3204045603533
// MI455X (gfx1250) — compile-verified
//
#include <hip/hip_runtime.h>

typedef __attribute__((ext_vector_type(2))) float v2f;
typedef __attribute__((ext_vector_type(8))) float v8f;

#define HD     64      // hidden dim
#define VO     64      // vocab
#define TWOH   128
#define SLOTS  8
#define BATCH  256
#define SEQL   2048
#define LN_EPS 1e-5f
#define THRESH 1.5f

// ---------------------------------------------------------------------------
// WMMA helpers (V_WMMA_F32_16X16X4_F32, full f32 precision, wave32).
// A fragment (16x4, M on lanes): lane<16 -> K = k0+0,k0+1 ; lane>=16 -> k0+2,k0+3
// B fragment (4x16, N on lanes): same K split, N = lane%16
// C/D: VGPR r holds row (r + 8*(lane>=16)), col = lane%16
// ---------------------------------------------------------------------------
__device__ inline v8f wmma_f32_tile(const float* __restrict__ A, int lda,
                                    const float* __restrict__ Bm, int ldb,
                                    int m0, int n0, int K, v8f c) {
  const int lane = threadIdx.x & 31;
  const int half = lane >> 4;
  const int mr   = lane & 15;
  for (int k0 = 0; k0 < K; k0 += 4) {
    const int ka = k0 + 2 * half;
    v2f a, b;
    a[0] = A[(m0 + mr) * lda + ka];
    a[1] = A[(m0 + mr) * lda + ka + 1];
    b[0] = Bm[(ka    ) * ldb + n0 + mr];
    b[1] = Bm[(ka + 1) * ldb + n0 + mr];
    c = __builtin_amdgcn_wmma_f32_16x16x4_f32(false, a, false, b, (short)0, c,
                                              false, false);
  }
  return c;
}

// Same but B is accessed transposed: B[k][n] = Bt[n][k]  (for h @ q_table^T)
__device__ inline v8f wmma_f32_tile_bt(const float* __restrict__ A, int lda,
                                       const float* __restrict__ Bt, int ldbt,
                                       int m0, int n0, int K, v8f c) {
  const int lane = threadIdx.x & 31;
  const int half = lane >> 4;
  const int mr   = lane & 15;
  for (int k0 = 0; k0 < K; k0 += 4) {
    const int ka = k0 + 2 * half;
    v2f a, b;
    a[0] = A[(m0 + mr) * lda + ka];
    a[1] = A[(m0 + mr) * lda + ka + 1];
    b[0] = Bt[(n0 + mr) * ldbt + ka];
    b[1] = Bt[(n0 + mr) * ldbt + ka + 1];
    c = __builtin_amdgcn_wmma_f32_16x16x4_f32(false, a, false, b, (short)0, c,
                                              false, false);
  }
  return c;
}

__device__ inline void store_tile(float* __restrict__ D, int ldd,
                                  int m0, int n0, v8f c) {
  const int lane = threadIdx.x & 31;
  const int half = lane >> 4;
  const int n    = n0 + (lane & 15);
#pragma unroll
  for (int r = 0; r < 8; ++r) D[(m0 + r + 8 * half) * ldd + n] = c[r];
}

// ---------------------------------------------------------------------------
// Kernel 1: single workgroup (8 waves). Precomputes all id-indexed tables:
//   h_table  = LN(E + relu(E@w1+b1)@w2 + b2)            [64 x 64]  (LDS)
//   score    = h_table @ gate_w + gate_b                [64]       -> order[]
//   q_table  = h_table @ q_w + q_b                      [64 x 64]  (LDS)
//   M        = h_table @ q_table^T / sqrt(H)            [64 x 64]  -> ws
//   out_tab  = h_table @ out_w                          [64 x 64]  -> ws
// ---------------------------------------------------------------------------
__global__ __launch_bounds__(256) void precompute_kernel(
    const float* __restrict__ embed, const float* __restrict__ w1,
    const float* __restrict__ b1,    const float* __restrict__ w2,
    const float* __restrict__ b2,    const float* __restrict__ ln_g,
    const float* __restrict__ ln_b,  const float* __restrict__ gate_w,
    const float* __restrict__ gate_b,const float* __restrict__ q_w,
    const float* __restrict__ q_b,   const float* __restrict__ out_w,
    float* __restrict__ ws_M, float* __restrict__ ws_out_table,
    int* __restrict__ ws_order) {
  __shared__ float smem[TWOH * HD + HD * HD + VO];  // 8192 + 4096 + 64 floats
  float* T1    = smem;                 // [64][128]; reused as q_table [64][64]
  float* h     = smem + TWOH * HD;     // [64][64]
  float* score = smem + TWOH * HD + HD * HD;
  const int wave = threadIdx.x >> 5;
  const int lane = threadIdx.x & 31;
  const int half = lane >> 4;
  const int mr   = lane & 15;

  // Stage 1: T1 = relu(E @ w1 + b1)   (4x8 tiles, 4 per wave)
  for (int t = wave * 4; t < wave * 4 + 4; ++t) {
    const int mt = t >> 3, nt = t & 7;
    v8f c = {};
    c = wmma_f32_tile(embed, HD, w1, TWOH, mt * 16, nt * 16, HD, c);
    const int n = nt * 16 + mr;
    const float bias = b1[n];
#pragma unroll
    for (int r = 0; r < 8; ++r) {
      float v = c[r] + bias;
      T1[(mt * 16 + r + 8 * half) * TWOH + n] = v > 0.f ? v : 0.f;
    }
  }
  __syncthreads();

  // Stage 2: z = E + T1 @ w2 + b2  (pre-LN, into h; 4x4 tiles, 2 per wave)
  for (int t = wave * 2; t < wave * 2 + 2; ++t) {
    const int mt = t >> 2, nt = t & 3;
    v8f c = {};
    c = wmma_f32_tile(T1, TWOH, w2, HD, mt * 16, nt * 16, TWOH, c);
    const int n = nt * 16 + mr;
    const float bias = b2[n];
#pragma unroll
    for (int r = 0; r < 8; ++r) {
      const int m = mt * 16 + r + 8 * half;
      h[m * HD + n] = c[r] + bias + embed[m * HD + n];
    }
  }
  __syncthreads();

  // LayerNorm per row (in place) + gate score. One row per thread (64 rows).
  if (threadIdx.x < VO) {
    const int r = threadIdx.x;
    float mu = 0.f;
    for (int k = 0; k < HD; ++k) mu += h[r * HD + k];
    mu *= (1.0f / HD);
    float var = 0.f;
    for (int k = 0; k < HD; ++k) { float d = h[r * HD + k] - mu; var += d * d; }
    var *= (1.0f / HD);
    const float inv = rsqrtf(var + LN_EPS);
    float sc = 0.f;
    for (int k = 0; k < HD; ++k) {
      float hv = (h[r * HD + k] - mu) * inv * ln_g[k] + ln_b[k];
      h[r * HD + k] = hv;
      sc += hv * gate_w[k];
    }
    score[r] = sc + gate_b[0];
  }
  __syncthreads();

  // Rank ids by score descending (stable: ties broken by smaller id first).
  if (threadIdx.x < VO) {
    const int i = threadIdx.x;
    const float si = score[i];
    int rank = 0;
    for (int j = 0; j < VO; ++j) {
      const float sj = score[j];
      if (sj > si || (sj == si && j < i)) ++rank;
    }
    ws_order[rank] = i;
  }

  // Stage 3a: q_table = h @ q_w + q_b (into T1 space);  out_table = h @ out_w
  float* qt = T1;  // T1 dead after stage-2 barrier
  for (int t = wave * 4; t < wave * 4 + 4; ++t) {
    const int which = t >> 4;  // 0 -> q_table, 1 -> out_table
    const int tt = t & 15;
    const int mt = tt >> 2, nt = tt & 3;
    v8f c = {};
    if (which == 0) {
      c = wmma_f32_tile(h, HD, q_w, HD, mt * 16, nt * 16, HD, c);
      const int n = nt * 16 + mr;
      const float bias = q_b[n];
#pragma unroll
      for (int r = 0; r < 8; ++r)
        qt[(mt * 16 + r + 8 * half) * HD + n] = c[r] + bias;
    } else {
      c = wmma_f32_tile(h, HD, out_w, HD, mt * 16, nt * 16, HD, c);
      store_tile(ws_out_table, HD, mt * 16, nt * 16, c);
    }
  }
  __syncthreads();

  // Stage 3b: M = h @ qt^T * (1/sqrt(H))   -> ws_M
  for (int t = wave * 2; t < wave * 2 + 2; ++t) {
    const int mt = t >> 2, nt = t & 3;
    v8f c = {};
    c = wmma_f32_tile_bt(h, HD, qt, HD, mt * 16, nt * 16, HD, c);
    const int n = nt * 16 + mr;
#pragma unroll
    for (int r = 0; r < 8; ++r)
      ws_M[(mt * 16 + r + 8 * half) * HD + n] = c[r] * 0.125f;
  }
}

// ---------------------------------------------------------------------------
// Kernel 2: one workgroup per batch row. LDS histogram of token ids (the only
// O(B*L) traffic in the whole problem: 2 MB total), slot selection by
// precomputed score order, 8-wide softmax + entropy gate, 64-wide output.
// ---------------------------------------------------------------------------
__global__ __launch_bounds__(256) void batch_kernel(
    const int* __restrict__ seq, const float* __restrict__ ws_M,
    const float* __restrict__ ws_out_table, const int* __restrict__ ws_order,
    const float* __restrict__ out_b, float* __restrict__ logits,
    float* __restrict__ ws_entropy) {
  __shared__ int   hist[VO];
  __shared__ int   slot_id[SLOTS];
  __shared__ float eff[SLOTS];
  const int b   = blockIdx.x;
  const int tid = threadIdx.x;

  if (tid < VO) hist[tid] = 0;
  __syncthreads();

  const int4* row = (const int4*)(seq + (size_t)b * SEQL);
  for (int i = tid; i < SEQL / 4; i += 256) {
    int4 v = row[i];
    atomicAdd(&hist[v.x], 1);
    atomicAdd(&hist[v.y], 1);
    atomicAdd(&hist[v.z], 1);
    atomicAdd(&hist[v.w], 1);
  }
  __syncthreads();

  if (tid == 0) {
    // Select 8 slots: ids in descending score order, bounded by occurrence.
    int rem = SLOTS, s = 0;
    for (int i = 0; i < VO && rem > 0; ++i) {
      const int id = ws_order[i];
      int c = hist[id];
      if (c > rem) c = rem;
      for (int k = 0; k < c; ++k) slot_id[s++] = id;
      rem -= c;
    }
    const int last_id = seq[(size_t)b * SEQL + (SEQL - 1)];
    float lg[SLOTS];
    float mx = -1e30f;
#pragma unroll
    for (int k = 0; k < SLOTS; ++k) {
      lg[k] = ws_M[slot_id[k] * HD + last_id];
      mx = fmaxf(mx, lg[k]);
    }
    float se = 0.f;
#pragma unroll
    for (int k = 0; k < SLOTS; ++k) { lg[k] = expf(lg[k] - mx); se += lg[k]; }
    const float inv = 1.f / se;
    float ent = 0.f;
#pragma unroll
    for (int k = 0; k < SLOTS; ++k) {
      const float a = lg[k] * inv;
      lg[k] = a;
      ent -= a * logf(a + 1e-9f);
    }
    ws_entropy[b] = ent;
    const float hi = (ent > THRESH) ? 1.f : 0.f;
#pragma unroll
    for (int k = 0; k < SLOTS; ++k)
      eff[k] = (1.f - hi) * lg[k] + hi * (1.f / SLOTS);
  }
  __syncthreads();

  if (tid < VO) {
    float acc = out_b[tid];
#pragma unroll
    for (int k = 0; k < SLOTS; ++k)
      acc += eff[k] * ws_out_table[slot_id[k] * HD + tid];
    logits[(size_t)b * VO + tid] = acc;
  }
}

// ---------------------------------------------------------------------------
// Kernel 3: mean of per-batch entropies -> d_out[B*V]
// ---------------------------------------------------------------------------
__global__ __launch_bounds__(256) void entropy_mean_kernel(
    const float* __restrict__ ws_entropy, float* __restrict__ out_scalar) {
  __shared__ float red[256];
  red[threadIdx.x] = ws_entropy[threadIdx.x];
  __syncthreads();
  for (int s = 128; s > 0; s >>= 1) {
    if (threadIdx.x < s) red[threadIdx.x] += red[threadIdx.x + s];
    __syncthreads();
  }
  if (threadIdx.x == 0) out_scalar[0] = red[0] * (1.0f / BATCH);
}

extern "C" void kernel_launch(void* const* d_in, const int* in_sizes, int n_in,
                              void* d_out, int out_size, void* d_ws,
                              size_t ws_size, hipStream_t stream) {
  const int*   seq    = (const int*)d_in[0];
  const float* embed  = (const float*)d_in[1];
  const float* w1     = (const float*)d_in[2];
  const float* b1     = (const float*)d_in[3];
  const float* w2     = (const float*)d_in[4];
  const float* b2     = (const float*)d_in[5];
  const float* ln_g   = (const float*)d_in[6];
  const float* ln_b   = (const float*)d_in[7];
  const float* gate_w = (const float*)d_in[8];
  const float* gate_b = (const float*)d_in[9];
  const float* q_w    = (const float*)d_in[10];
  const float* q_b    = (const float*)d_in[11];
  const float* out_w  = (const float*)d_in[12];
  const float* out_b  = (const float*)d_in[13];
  (void)in_sizes; (void)n_in; (void)out_size; (void)ws_size;

  float* logits   = (float*)d_out;              // [256 * 64]
  float* ent_mean = (float*)d_out + BATCH * VO; // [1]

  float* ws           = (float*)d_ws;
  float* ws_M         = ws;                         // 4096 floats
  float* ws_out_table = ws + HD * HD;               // 4096 floats
  int*   ws_order     = (int*)(ws + 2 * HD * HD);   // 64 ints
  float* ws_entropy   = ws + 2 * HD * HD + VO;      // 256 floats

  precompute_kernel<<<1, 256, 0, stream>>>(embed, w1, b1, w2, b2, ln_g, ln_b,
                                           gate_w, gate_b, q_w, q_b, out_w,
                                           ws_M, ws_out_table, ws_order);
  batch_kernel<<<BATCH, 256, 0, stream>>>(seq, ws_M, ws_out_table, ws_order,
                                          out_b, logits, ws_entropy);
  entropy_mean_kernel<<<1, 256, 0, stream>>>(ws_entropy, ent_mean);
}